// GNNLocalizationModel_45509473468553
// MI455X (gfx1250) — compile-verified
//
#include <hip/hip_runtime.h>
#include <math.h>

#define NNODES 40000
#define NEDGES 320000
#define ENL    (NEDGES + NNODES)   // edges including self-loops
#define NG     400

typedef __attribute__((ext_vector_type(2))) float v2f;
typedef __attribute__((ext_vector_type(8))) float v8f;

// ---------------- layer 0 input GEMM: [N,3] @ [3,512] ----------------
__global__ void k_gemm0(const float* __restrict__ x, const float* __restrict__ W,
                        float* __restrict__ out) {
    int t = blockIdx.x * blockDim.x + threadIdx.x;   // N*512 threads, exact
    int n = t >> 9, f = t & 511;
    out[t] = x[n * 3 + 0] * W[f] + x[n * 3 + 1] * W[512 + f] + x[n * 3 + 2] * W[1024 + f];
}

// ---------------- fp32 WMMA GEMM: A[M,K] @ B[K,Ncol], one 16x16 tile / wave ----------------
__global__ void k_gemm_wmma(const float* __restrict__ A, const float* __restrict__ B,
                            float* __restrict__ C, int M, int K, int Ncol) {
    int wave = (blockIdx.x * blockDim.x + threadIdx.x) >> 5;
    int lane = threadIdx.x & 31;
    int ntiles = Ncol >> 4;
    int mt = wave / ntiles;
    int nt = wave - mt * ntiles;
    int m0 = mt << 4, n0 = nt << 4;
    if (m0 >= M) return;                       // wave-uniform: EXEC stays all-1s
    int l  = lane & 15;
    int kh = (lane >> 4) << 1;                 // lanes 0-15 -> K+0/1, lanes 16-31 -> K+2/3
    const float* ap = A + (size_t)(m0 + l) * K + kh;
    const float* bp = B + (size_t)kh * Ncol + (n0 + l);
    v8f acc = {0.f, 0.f, 0.f, 0.f, 0.f, 0.f, 0.f, 0.f};
    #pragma unroll 4
    for (int k = 0; k < K; k += 4) {
        v2f a, b;
        a[0] = ap[k];
        a[1] = ap[k + 1];
        b[0] = bp[(size_t)k * Ncol];
        b[1] = bp[(size_t)(k + 1) * Ncol];
        acc = __builtin_amdgcn_wmma_f32_16x16x4_f32(false, a, false, b,
                                                    (short)0, acc, false, false);
    }
    int rbase = m0 + ((lane >> 4) << 3);       // C/D layout: lanes16-31 hold rows M=8..15
    #pragma unroll
    for (int v = 0; v < 8; ++v)
        C[(size_t)(rbase + v) * Ncol + n0 + l] = acc[v];
}

// ---------------- attention coefficients: al_s/al_d [N,H] ----------------
__global__ void k_alpha_coef(const float* __restrict__ Hm, const float* __restrict__ as_,
                             const float* __restrict__ ad_, float* __restrict__ als,
                             float* __restrict__ ald, int H, int Cc, int F) {
    int t = blockIdx.x * blockDim.x + threadIdx.x;
    if (t >= NNODES * H) return;
    int n = t / H, h = t - n * H;
    const float* hp  = Hm + (size_t)n * F + h * Cc;
    const float* asp = as_ + h * Cc;
    const float* adp = ad_ + h * Cc;
    float s = 0.f, d = 0.f;
    for (int c = 0; c < Cc; ++c) { float v = hp[c]; s += v * asp[c]; d += v * adp[c]; }
    als[t] = s;
    ald[t] = d;
}

// ---------------- CSR-by-dst construction (once per call) ----------------
__global__ void k_csr_count(const int* __restrict__ ei, int* __restrict__ deg) {
    int e = blockIdx.x * blockDim.x + threadIdx.x;
    if (e >= ENL) return;
    int dst = (e < NEDGES) ? ei[NEDGES + e] : (e - NEDGES);
    atomicAdd(&deg[dst], 1);
}

#define SCAN_T 1024
__global__ void k_csr_scan(const int* __restrict__ deg, int* __restrict__ rowstart) {
    __shared__ int part[SCAN_T];
    int t = threadIdx.x;
    const int CH = (NNODES + SCAN_T - 1) / SCAN_T;   // 40 elems / thread
    int base = t * CH;
    int s = 0;
    for (int i = 0; i < CH; ++i) { int idx = base + i; if (idx < NNODES) s += deg[idx]; }
    part[t] = s;
    __syncthreads();
    if (t == 0) {
        int run = 0;
        for (int i = 0; i < SCAN_T; ++i) { int v = part[i]; part[i] = run; run += v; }
    }
    __syncthreads();
    int run = part[t];
    for (int i = 0; i < CH; ++i) {
        int idx = base + i;
        if (idx < NNODES) { rowstart[idx] = run; run += deg[idx]; }
    }
    if (t == SCAN_T - 1) rowstart[NNODES] = run;     // == ENL
}

__global__ void k_csr_fill(const int* __restrict__ ei, const int* __restrict__ rowstart,
                           int* __restrict__ cursor, int* __restrict__ elist) {
    int e = blockIdx.x * blockDim.x + threadIdx.x;
    if (e >= ENL) return;
    int dst = (e < NEDGES) ? ei[NEDGES + e] : (e - NEDGES);
    int pos = atomicAdd(&cursor[dst], 1);
    elist[rowstart[dst] + pos] = e;
}

// ---- fused segment-softmax + message aggregation: one wave per (dst node, head) ----
// No float atomics: each wave owns its output row slice. Cc is always 64 (2 regs/lane).
__global__ void k_gat_fused(const float* __restrict__ als, const float* __restrict__ ald,
                            const int* __restrict__ ei, const int* __restrict__ rowstart,
                            const int* __restrict__ elist, const float* __restrict__ Hm,
                            float* __restrict__ agg, int H, int F) {
    int gid  = blockIdx.x * blockDim.x + threadIdx.x;
    int w    = gid >> 5;
    int lane = gid & 31;
    if (w >= NNODES * H) return;               // grids are exact; wave-uniform anyway
    int n = w / H, h = w - n * H;
    int beg = rowstart[n], end = rowstart[n + 1];
    float aldn = ald[n * H + h];

    // pass 1: segment max of leaky-relu logits (lane-parallel over edges)
    float mx = -3.402823466e38f;
    for (int i = beg + lane; i < end; i += 32) {
        int e   = elist[i];
        int src = (e < NEDGES) ? ei[e] : (e - NEDGES);
        float v = als[src * H + h] + aldn;
        v = v > 0.f ? v : 0.2f * v;
        mx = fmaxf(mx, v);
    }
    #pragma unroll
    for (int off = 16; off > 0; off >>= 1)
        mx = fmaxf(mx, __shfl_xor(mx, off, 32));

    // pass 2: accumulate exp-weighted messages (lane-parallel over 64 channels)
    float acc0 = 0.f, acc1 = 0.f, den = 0.f;
    for (int i = beg; i < end; ++i) {
        int e   = elist[i];                              // broadcast (scalar) loads
        int src = (e < NEDGES) ? ei[e] : (e - NEDGES);
        float v = als[src * H + h] + aldn;
        v = v > 0.f ? v : 0.2f * v;
        float ex = expf(v - mx);
        den += ex;
        const float* hs = Hm + (size_t)src * F + h * 64; // 256B coalesced gather
        acc0 += ex * hs[lane];
        acc1 += ex * hs[lane + 32];
    }
    float inv = 1.0f / den;                              // self-loop guarantees den > 0
    float* ag = agg + (size_t)n * F + h * 64;
    ag[lane]      = acc0 * inv;
    ag[lane + 32] = acc1 * inv;
}

// ---------------- bias + eval-BN + ReLU (+ optional residual) ----------------
__global__ void k_finalize(const float* __restrict__ agg, const float* __restrict__ bias,
                           const float* __restrict__ g, const float* __restrict__ be,
                           const float* __restrict__ res, float* __restrict__ out,
                           int F, int total) {
    int t = blockIdx.x * blockDim.x + threadIdx.x;
    if (t >= total) return;
    int f = t % F;
    float sc = g[f] * rsqrtf(1.0f + 1e-5f);
    float v  = (agg[t] + bias[f]) * sc + be[f];
    v = v > 0.f ? v : 0.f;
    if (res) v += res[t];
    out[t] = v;
}

// ---------------- contiguous per-graph mean/max pooling (100 nodes/graph, 64 ch) -------
__global__ void k_pool(const float* __restrict__ x3, float* __restrict__ meanp,
                       float* __restrict__ maxp) {
    int g = blockIdx.x, c = threadIdx.x;     // 64 threads
    const float* base = x3 + (size_t)g * 100 * 64 + c;
    float s = 0.f, mx = -3.402823466e38f;
    for (int n = 0; n < 100; ++n) { float v = base[n * 64]; s += v; mx = fmaxf(mx, v); }
    meanp[g * 64 + c] = s / 100.0f;
    maxp[g * 64 + c]  = mx;
}

// ---------------- fused regressor head: one block per graph ----------------
__global__ void k_head(const float* __restrict__ meanp, const float* __restrict__ maxp,
                       const float* __restrict__ gfeat,
                       const float* __restrict__ mlp_w, const float* __restrict__ mlp_b,
                       const float* __restrict__ w1, const float* __restrict__ b1,
                       const float* __restrict__ g1, const float* __restrict__ be1,
                       const float* __restrict__ w2, const float* __restrict__ b2,
                       const float* __restrict__ g2, const float* __restrict__ be2,
                       const float* __restrict__ w3, const float* __restrict__ b3,
                       float* __restrict__ out) {
    __shared__ float z[160], r1[64], r2[32];
    int g = blockIdx.x, t = threadIdx.x;     // 64 threads
    z[t]      = meanp[g * 64 + t];
    z[64 + t] = maxp[g * 64 + t];
    if (t < 32) {
        float s = mlp_b[t];
        for (int k = 0; k < 3; ++k) s += gfeat[g * 3 + k] * mlp_w[k * 32 + t];
        z[128 + t] = s > 0.f ? s : 0.f;
    }
    __syncthreads();
    {
        float s = b1[t];
        for (int k = 0; k < 160; ++k) s += z[k] * w1[k * 64 + t];
        s = s * (g1[t] * rsqrtf(1.0f + 1e-5f)) + be1[t];
        r1[t] = s > 0.f ? s : 0.f;
    }
    __syncthreads();
    if (t < 32) {
        float s = b2[t];
        for (int k = 0; k < 64; ++k) s += r1[k] * w2[k * 32 + t];
        s = s * (g2[t] * rsqrtf(1.0f + 1e-5f)) + be2[t];
        r2[t] = s > 0.f ? s : 0.f;
    }
    __syncthreads();
    if (t < 2) {
        float s = b3[t];
        for (int k = 0; k < 32; ++k) s += r2[k] * w3[k * 2 + t];
        out[g * 2 + t] = s;
    }
}

extern "C" void kernel_launch(void* const* d_in, const int* in_sizes, int n_in,
                              void* d_out, int out_size, void* d_ws, size_t ws_size,
                              hipStream_t stream) {
    const float* x     = (const float*)d_in[0];
    const int*   ei    = (const int*)d_in[1];
    // d_in[2] = batch: contiguous repeat(arange(G), 100) by construction
    const float* gfeat = (const float*)d_in[3];
    const float* W0  = (const float*)d_in[4];
    const float* as0 = (const float*)d_in[5];
    const float* ad0 = (const float*)d_in[6];
    const float* b0  = (const float*)d_in[7];
    const float* g0  = (const float*)d_in[8];
    const float* be0 = (const float*)d_in[9];
    const float* W1  = (const float*)d_in[10];
    const float* as1 = (const float*)d_in[11];
    const float* ad1 = (const float*)d_in[12];
    const float* b1  = (const float*)d_in[13];
    const float* g1  = (const float*)d_in[14];
    const float* be1 = (const float*)d_in[15];
    const float* W2  = (const float*)d_in[16];
    const float* as2 = (const float*)d_in[17];
    const float* ad2 = (const float*)d_in[18];
    const float* b2  = (const float*)d_in[19];
    const float* g2  = (const float*)d_in[20];
    const float* be2 = (const float*)d_in[21];
    const float* mlp_w = (const float*)d_in[22];
    const float* mlp_b = (const float*)d_in[23];
    const float* rw1 = (const float*)d_in[24];
    const float* rb1 = (const float*)d_in[25];
    const float* rg1 = (const float*)d_in[26];
    const float* rbe1= (const float*)d_in[27];
    const float* rw2 = (const float*)d_in[28];
    const float* rb2 = (const float*)d_in[29];
    const float* rg2 = (const float*)d_in[30];
    const float* rbe2= (const float*)d_in[31];
    const float* rw3 = (const float*)d_in[32];
    const float* rb3 = (const float*)d_in[33];

    float* ws = (float*)d_ws;
    const size_t NS = (size_t)NNODES * 512;
    float* Hm    = ws;                       // layer GEMM output; reused as x2 after layer 1
    float* agg   = ws + NS;                  // fused-kernel output (pre-bias/BN)
    float* x1    = ws + 2 * NS;              // layer-0 output; first N*64 reused as Hmat2
    float* sm    = ws + 3 * NS;              // small buffers
    float* als   = sm;                       // [N,8]
    float* ald   = sm + (size_t)NNODES * 8;  // [N,8]
    float* meanp = sm + (size_t)2 * NNODES * 8;
    float* maxp  = meanp + NG * 64;
    int*   rowstart = (int*)(maxp + NG * 64);        // [N+1]
    int*   deg      = rowstart + NNODES + 1;         // [N], reused as scatter cursor
    int*   elist    = deg + NNODES;                  // [ENL]
    float* Hm2 = x1;                                 // [N,64]
    float* x3  = x1 + (size_t)NNODES * 64;           // [N,64]
    float* x2  = Hm;                                 // [N,512]

    // ================= CSR by destination (built once, reused by all 3 layers) ==========
    (void)hipMemsetAsync(deg, 0, (size_t)NNODES * sizeof(int), stream);
    k_csr_count<<<(ENL + 255) / 256, 256, 0, stream>>>(ei, deg);
    k_csr_scan<<<1, SCAN_T, 0, stream>>>(deg, rowstart);
    (void)hipMemsetAsync(deg, 0, (size_t)NNODES * sizeof(int), stream);   // now cursor
    k_csr_fill<<<(ENL + 255) / 256, 256, 0, stream>>>(ei, rowstart, deg, elist);

    // ================= Layer 0: GAT(3 -> 8x64) =================
    k_gemm0<<<(NNODES * 512) / 256, 256, 0, stream>>>(x, W0, Hm);
    k_alpha_coef<<<(NNODES * 8 + 255) / 256, 256, 0, stream>>>(Hm, as0, ad0, als, ald, 8, 64, 512);
    k_gat_fused<<<(NNODES * 8) / 8, 256, 0, stream>>>(als, ald, ei, rowstart, elist, Hm, agg, 8, 512);
    k_finalize<<<((int)NS + 255) / 256, 256, 0, stream>>>(agg, b0, g0, be0, nullptr, x1, 512, (int)NS);

    // ================= Layer 1: GAT(512 -> 8x64), residual =================
    k_gemm_wmma<<<(2500 * 32) / 8, 256, 0, stream>>>(x1, W1, Hm, NNODES, 512, 512);
    k_alpha_coef<<<(NNODES * 8 + 255) / 256, 256, 0, stream>>>(Hm, as1, ad1, als, ald, 8, 64, 512);
    k_gat_fused<<<(NNODES * 8) / 8, 256, 0, stream>>>(als, ald, ei, rowstart, elist, Hm, agg, 8, 512);
    k_finalize<<<((int)NS + 255) / 256, 256, 0, stream>>>(agg, b1, g1, be1, x1, x2, 512, (int)NS);

    // ================= Layer 2: GAT(512 -> 1x64) =================
    k_gemm_wmma<<<(2500 * 4) / 8, 256, 0, stream>>>(x2, W2, Hm2, NNODES, 512, 64);
    k_alpha_coef<<<(NNODES + 255) / 256, 256, 0, stream>>>(Hm2, as2, ad2, als, ald, 1, 64, 64);
    k_gat_fused<<<NNODES / 8, 256, 0, stream>>>(als, ald, ei, rowstart, elist, Hm2, agg, 1, 64);
    k_finalize<<<(NNODES * 64 + 255) / 256, 256, 0, stream>>>(agg, b2, g2, be2, nullptr, x3, 64, NNODES * 64);

    // ================= Pooling + regressor head =================
    k_pool<<<NG, 64, 0, stream>>>(x3, meanp, maxp);
    k_head<<<NG, 64, 0, stream>>>(meanp, maxp, gfeat, mlp_w, mlp_b,
                                  rw1, rb1, rg1, rbe1, rw2, rb2, rg2, rbe2, rw3, rb3,
                                  (float*)d_out);
}